// ISDALoss_EM_19619410608755
// MI455X (gfx1250) — compile-verified
//
#include <hip/hip_runtime.h>
#include <hip/hip_bf16.h>
#include <math.h>

typedef float v2f __attribute__((ext_vector_type(2)));
typedef float v8f __attribute__((ext_vector_type(8)));

#define NSAMP 2048
#define NCLS  100
#define NCLSP 112   // padded to 7*16
#define NK    4
#define NA    128
#define NCK   (NCLS * NK)          // 400
#define MATSZ (NA * NA)            // 16384
#define LDSP  129                  // LDS row stride (bank-conflict pad)

// ---------------- workspace layout (floats) ----------------
#define WS_L      0
#define WS_LINV   (WS_L    + (size_t)NCK * MATSZ)      // 6,553,600
#define WS_LOGDET (WS_LINV + (size_t)NCK * MATSZ)      // +400
#define WS_WP     (WS_LOGDET + NCK)                    // 112*128
#define WS_Y      (WS_WP   + (size_t)NCLSP * NA)       // 400*112*128
#define WS_KSTAR  (WS_Y    + (size_t)NCK * NCLSP * NA) // 2048 ints
#define WS_LOSSN  (WS_KSTAR + NSAMP)                   // 2048

// ============================================================
// K1: per-(c,k) Cholesky + logdet + triangular inverse (LDS)
// ============================================================
__global__ void chol_trinv_kernel(const float* __restrict__ sigma,
                                  float* __restrict__ Lg,
                                  float* __restrict__ Linvg,
                                  float* __restrict__ logdetg) {
  extern __shared__ float sm[];
  float* Am  = sm;               // [128][129]
  float* Inv = sm + NA * LDSP;   // [128][129]
  const int ck  = blockIdx.x;
  const int tid = threadIdx.x;   // 0..127, owns row tid (chol) / col tid (inv)
  const float* S = sigma + (size_t)ck * MATSZ;

  for (int j = 0; j < NA; ++j) Am[tid * LDSP + j] = S[tid * NA + j];
  __syncthreads();

  // right-looking Cholesky, lower triangle of Am
  for (int j = 0; j < NA; ++j) {
    if (tid == j) Am[j * LDSP + j] = sqrtf(Am[j * LDSP + j]);
    __syncthreads();
    float djj = Am[j * LDSP + j];
    if (tid > j) Am[tid * LDSP + j] /= djj;
    __syncthreads();
    if (tid > j) {
      float lij = Am[tid * LDSP + j];
      for (int l = j + 1; l <= tid; ++l)
        Am[tid * LDSP + l] -= lij * Am[l * LDSP + j];
    }
    __syncthreads();
  }

  // logdet = 2 * sum(log L_ii)   (use Inv row 0 as scratch; safe, see below)
  Inv[tid] = logf(Am[tid * LDSP + tid]);
  __syncthreads();
  for (int off = 64; off > 0; off >>= 1) {
    if (tid < off) Inv[tid] += Inv[tid + off];
    __syncthreads();
  }
  if (tid == 0) logdetg[ck] = 2.0f * Inv[0];
  __syncthreads();

  // store L with zeroed upper triangle
  for (int j = 0; j < NA; ++j)
    Lg[(size_t)ck * MATSZ + tid * NA + j] = (j <= tid) ? Am[tid * LDSP + j] : 0.0f;

  // triangular inverse: thread tid computes column j=tid (columns independent)
  {
    const int j = tid;
    Inv[j * LDSP + j] = 1.0f / Am[j * LDSP + j];
    for (int i = j + 1; i < NA; ++i) {
      float s = 0.0f;
      for (int t = j; t < i; ++t) s += Am[i * LDSP + t] * Inv[t * LDSP + j];
      Inv[i * LDSP + j] = -s / Am[i * LDSP + i];
    }
  }
  __syncthreads();
  for (int j = 0; j < NA; ++j)
    Linvg[(size_t)ck * MATSZ + tid * NA + j] = (j <= tid) ? Inv[tid * LDSP + j] : 0.0f;
}

// ============================================================
// K2: zero-pad weight [100,128] -> Wp [112,128]
// ============================================================
__global__ void pad_weight_kernel(const float* __restrict__ W, float* __restrict__ Wp) {
  int i = blockIdx.x * blockDim.x + threadIdx.x;   // over 112*128
  if (i < NCLSP * NA) {
    int c = i / NA;
    Wp[i] = (c < NCLS) ? W[i] : 0.0f;
  }
}

// ============================================================
// K3: Y[ck] = Wp(112x128) @ L[ck](128x128), WMMA f32 16x16x4
//     one block per ck, 8 waves x 7 tiles (7 Mtiles x 8 Ntiles)
// ============================================================
__global__ void y_of_L_wmma_kernel(const float* __restrict__ Wp,
                                   const float* __restrict__ Lg,
                                   float* __restrict__ Yg) {
  const int ck    = blockIdx.x;
  const int wave  = threadIdx.x >> 5;
  const int lane  = threadIdx.x & 31;
  const int half  = lane >> 4;        // 0 or 1
  const int l16   = lane & 15;
  const float* Lck = Lg + (size_t)ck * MATSZ;
  float*       Yck = Yg + (size_t)ck * NCLSP * NA;

  for (int t = 0; t < 7; ++t) {
    const int tile = wave * 7 + t;          // 0..55
    const int m0 = (tile % 7) * 16;         // row tile in Wp
    const int n0 = (tile / 7) * 16;         // col tile in L
    const int m  = m0 + l16;
    v8f acc = {};
    for (int kk = 0; kk < 32; ++kk) {
      const int kb = kk * 4;
      v2f a, b;
      a.x = Wp[m * NA + kb + 2 * half];
      a.y = Wp[m * NA + kb + 2 * half + 1];
      b.x = Lck[(kb + 2 * half) * NA + n0 + l16];
      b.y = Lck[(kb + 2 * half + 1) * NA + n0 + l16];
      acc = __builtin_amdgcn_wmma_f32_16x16x4_f32(false, a, false, b,
                                                  (short)0, acc, false, false);
    }
    for (int r = 0; r < 8; ++r) {
      const int row = m0 + r + 8 * half;
      Yck[(size_t)row * NA + n0 + l16] = acc[r];
    }
  }
}

// ============================================================
// K4: y = features(2048x128) @ Wp^T + bias -> d_out[1 + n*100 + c]
//     896 output tiles, 8 waves/block -> 112 blocks
// ============================================================
__global__ void fc_wmma_kernel(const float* __restrict__ F,
                               const float* __restrict__ Wp,
                               const float* __restrict__ bias,
                               float* __restrict__ out) {
  const int wave = threadIdx.x >> 5;
  const int lane = threadIdx.x & 31;
  const int half = lane >> 4;
  const int l16  = lane & 15;
  const int tile = blockIdx.x * 8 + wave;   // 0..895
  const int m0 = (tile / 7) * 16;           // sample tile
  const int n0 = (tile % 7) * 16;           // class tile
  const int m  = m0 + l16;
  const int n  = n0 + l16;

  v8f acc = {};
  for (int kk = 0; kk < 32; ++kk) {
    const int kb = kk * 4;
    v2f a, b;
    a.x = F[(size_t)m * NA + kb + 2 * half];
    a.y = F[(size_t)m * NA + kb + 2 * half + 1];
    b.x = Wp[n * NA + kb + 2 * half];       // B[k][n] = Wp[n][k]
    b.y = Wp[n * NA + kb + 2 * half + 1];
    acc = __builtin_amdgcn_wmma_f32_16x16x4_f32(false, a, false, b,
                                                (short)0, acc, false, false);
  }
  for (int r = 0; r < 8; ++r) {
    const int row = m0 + r + 8 * half;
    const int c   = n0 + l16;
    if (c < NCLS)
      out[1 + (size_t)row * NCLS + c] = acc[r] + bias[c];
  }
}

// ============================================================
// K5: per-sample k* = argmin_k ( ||Linv diff||^2 + logdet )
// ============================================================
__global__ void kstar_kernel(const float* __restrict__ F,
                             const float* __restrict__ mu,
                             const float* __restrict__ Linvg,
                             const float* __restrict__ logdetg,
                             const int* __restrict__ labels,
                             int* __restrict__ kstar) {
  __shared__ float diffs[NA];
  __shared__ float red[NA];
  const int n   = blockIdx.x;
  const int tid = threadIdx.x;
  const int l   = labels[n];
  float best = -3.0e38f;
  int   kst  = 0;
  for (int k = 0; k < NK; ++k) {
    const int ck = l * NK + k;
    diffs[tid] = F[(size_t)n * NA + tid] - mu[(size_t)ck * NA + tid];
    __syncthreads();
    const float* Li = Linvg + (size_t)ck * MATSZ + tid * NA;
    float s = 0.0f;
    for (int j = 0; j <= tid; ++j) s += Li[j] * diffs[j];   // lower triangular
    red[tid] = s * s;
    __syncthreads();
    for (int off = 64; off > 0; off >>= 1) {
      if (tid < off) red[tid] += red[tid + off];
      __syncthreads();
    }
    if (tid == 0) {
      float score = -(red[0] + logdetg[ck]);
      if (score > best) { best = score; kst = k; }
    }
    __syncthreads();
  }
  if (tid == 0) kstar[n] = kst;
}

// ============================================================
// K6: sigma2 from Y rows, aug = y + 0.5*ratio*||Yc - Yl||^2,
//     log-softmax at label -> lossn[n]
// ============================================================
__global__ void loss_kernel(const float* __restrict__ Yg,
                            const float* __restrict__ out_y,   // d_out (y at +1)
                            const int* __restrict__ labels,
                            const int* __restrict__ kstar,
                            const float* __restrict__ ratiop,
                            float* __restrict__ lossn) {
  __shared__ float Yl[NA];
  __shared__ float augs[NCLSP];
  const int n    = blockIdx.x;
  const int tid  = threadIdx.x;        // 256 threads = 8 waves
  const int wave = tid >> 5;
  const int lane = tid & 31;
  const int l    = labels[n];
  const int ck   = l * NK + kstar[n];
  const float ratio = ratiop[0];
  const float* Yck  = Yg + (size_t)ck * NCLSP * NA;
  const float* yrow = out_y + 1 + (size_t)n * NCLS;

  if (tid < NA) Yl[tid] = Yck[(size_t)l * NA + tid];
  __syncthreads();

  for (int c = wave; c < NCLS; c += 8) {
    const float* Yc = Yck + (size_t)c * NA;
    float s = 0.0f;
    for (int a = lane; a < NA; a += 32) {
      float d = Yc[a] - Yl[a];
      s += d * d;
    }
    for (int off = 16; off > 0; off >>= 1) s += __shfl_xor(s, off, 32);
    if (lane == 0) augs[c] = yrow[c] + 0.5f * ratio * s;
  }
  __syncthreads();

  if (tid == 0) {
    float m = -3.0e38f;
    for (int c = 0; c < NCLS; ++c) m = fmaxf(m, augs[c]);
    float se = 0.0f;
    for (int c = 0; c < NCLS; ++c) se += expf(augs[c] - m);
    lossn[n] = (m + logf(se)) - augs[l];   // -log_softmax at label
  }
}

// ============================================================
// K7: loss = mean(lossn) -> d_out[0]
// ============================================================
__global__ void reduce_loss_kernel(const float* __restrict__ lossn,
                                   float* __restrict__ out) {
  __shared__ float red[256];
  const int tid = threadIdx.x;
  float s = 0.0f;
  for (int i = tid; i < NSAMP; i += 256) s += lossn[i];
  red[tid] = s;
  __syncthreads();
  for (int off = 128; off > 0; off >>= 1) {
    if (tid < off) red[tid] += red[tid + off];
    __syncthreads();
  }
  if (tid == 0) out[0] = red[0] / (float)NSAMP;
}

// ============================================================
extern "C" void kernel_launch(void* const* d_in, const int* in_sizes, int n_in,
                              void* d_out, int out_size, void* d_ws, size_t ws_size,
                              hipStream_t stream) {
  const float* features = (const float*)d_in[0];   // [2048,128]
  const float* weight   = (const float*)d_in[1];   // [100,128]
  const float* bias     = (const float*)d_in[2];   // [100]
  // d_in[3] = pi  -- unused (q collapses to one-hot argmax)
  const float* mu       = (const float*)d_in[4];   // [100,4,128]
  const float* sigma    = (const float*)d_in[5];   // [100,4,128,128]
  const int*   labels   = (const int*)d_in[6];     // [2048]
  const float* ratiop   = (const float*)d_in[7];   // scalar

  float* out = (float*)d_out;                      // [0]=loss, [1..]=y[2048,100]
  float* ws  = (float*)d_ws;

  float* Lg      = ws + WS_L;
  float* Linvg   = ws + WS_LINV;
  float* logdetg = ws + WS_LOGDET;
  float* Wp      = ws + WS_WP;
  float* Yg      = ws + WS_Y;
  int*   kstar   = (int*)(ws + WS_KSTAR);
  float* lossn   = ws + WS_LOSSN;

  // K1: 400 Cholesky + triangular inverses; 2*128*129 floats of dynamic LDS
  chol_trinv_kernel<<<NCK, NA, 2 * NA * LDSP * sizeof(float), stream>>>(
      sigma, Lg, Linvg, logdetg);

  // K2: pad weights
  pad_weight_kernel<<<(NCLSP * NA + 255) / 256, 256, 0, stream>>>(weight, Wp);

  // K3: Y[ck] = Wp @ L[ck]  (WMMA)
  y_of_L_wmma_kernel<<<NCK, 256, 0, stream>>>(Wp, Lg, Yg);

  // K4: y = F @ Wp^T + bias  (WMMA) -> d_out
  fc_wmma_kernel<<<112, 256, 0, stream>>>(features, Wp, bias, out);

  // K5: per-sample cluster argmax
  kstar_kernel<<<NSAMP, NA, 0, stream>>>(features, mu, Linvg, logdetg, labels, kstar);

  // K6: sigma2 + aug + per-sample NLL
  loss_kernel<<<NSAMP, 256, 0, stream>>>(Yg, out, labels, kstar, ratiop, lossn);

  // K7: mean -> d_out[0]
  reduce_loss_kernel<<<1, 256, 0, stream>>>(lossn, out);
}